// Detector_57604101373957
// MI455X (gfx1250) — compile-verified
//
#include <hip/hip_runtime.h>

// GCN: N=100000 nodes, E=1600000 edges, 128 -> 64 -> 64 -> 2, log_softmax.
// - fp32 WMMA (v_wmma_f32_16x16x4_f32) for dense transforms (exact fp32 on matrix pipe)
// - L2-resident atomic scatter for edge aggregation (feature matrix 25.6MB << 192MB L2)
// - per-edge norm precomputed once; scatter does 2 features/thread (float2)
// - elementwise passes vectorized to b128

#define N_NODES 100000
#define E_EDGES 1600000
#define IN_DIM  128
#define HID     64

typedef __attribute__((ext_vector_type(2))) float v2f;
typedef __attribute__((ext_vector_type(8))) float v8f;

// ---------------- degree / norm kernels ----------------
__global__ void k_deg_init(float* __restrict__ deg, int n) {
    int i = blockIdx.x * blockDim.x + threadIdx.x;
    if (i < n) deg[i] = 1.0f;                       // self-loop contributes 1
}

__global__ void k_deg_accum(const int* __restrict__ dst, float* __restrict__ deg, int e) {
    int i = blockIdx.x * blockDim.x + threadIdx.x;
    if (i < e) atomicAdd(&deg[dst[i]], 1.0f);
}

__global__ void k_deg_finalize(float* __restrict__ deg, int n) {
    int i = blockIdx.x * blockDim.x + threadIdx.x;
    if (i < n) deg[i] = rsqrtf(deg[i]);             // deg >= 1 always (self-loops)
}

// norm[e] = dis[src[e]] * dis[dst[e]]  (computed once, reused by both layers)
__global__ void k_edge_norm(const int* __restrict__ src, const int* __restrict__ dst,
                            const float* __restrict__ dis, float* __restrict__ norm, int e) {
    int i = blockIdx.x * blockDim.x + threadIdx.x;
    if (i < e) norm[i] = dis[src[i]] * dis[dst[i]];
}

// ---------------- WMMA fp32 GEMM: C[M x 64] = A[M x K] @ W[K x 64] ----------------
// One wave computes a 16-row x 64-col strip with 4 v8f accumulators,
// stepping K by 4 using V_WMMA_F32_16X16X4_F32.
//
// fp32 A 16x4 layout (ISA 7.12.2): lanes 0-15 = M, VGPR0/1 = K0/K1;
//                                  lanes 16-31 = same M, K2/K3.
// fp32 B 4x16 layout (mirrored):   VGPR0 lanes 0-15 = row k, lanes 16-31 = row k+2;
//                                  VGPR1 = rows k+1 / k+3.
// fp32 C/D 16x16: VGPR v -> M = v (lanes 0-15) / v+8 (lanes 16-31), N = lane&15.
__global__ void k_gemm_wmma(const float* __restrict__ A, const float* __restrict__ W,
                            float* __restrict__ C, int M, int K) {
    const int lane = threadIdx.x & 31;
    const int wave = blockIdx.x * (blockDim.x >> 5) + (threadIdx.x >> 5);
    const long long row0 = (long long)wave * 16;
    if (row0 >= M) return;                           // wave-uniform; EXEC stays all-1s
    const int half = lane >> 4;                      // 0: K+0/1, 1: K+2/3
    const int l16  = lane & 15;
    const float* arow = A + (row0 + l16) * (long long)K;

    v8f acc0 = {}, acc1 = {}, acc2 = {}, acc3 = {};
    for (int k = 0; k < K; k += 4) {
        const int ka = k + (half << 1);
        v2f a = *(const v2f*)(arow + ka);            // A[r][ka], A[r][ka+1]
        const float* wr = W + ka * HID + l16;
        v2f b0 = { wr[ 0], wr[HID +  0] };
        v2f b1 = { wr[16], wr[HID + 16] };
        v2f b2 = { wr[32], wr[HID + 32] };
        v2f b3 = { wr[48], wr[HID + 48] };
        acc0 = __builtin_amdgcn_wmma_f32_16x16x4_f32(false, a, false, b0, (short)0, acc0, false, false);
        acc1 = __builtin_amdgcn_wmma_f32_16x16x4_f32(false, a, false, b1, (short)0, acc1, false, false);
        acc2 = __builtin_amdgcn_wmma_f32_16x16x4_f32(false, a, false, b2, (short)0, acc2, false, false);
        acc3 = __builtin_amdgcn_wmma_f32_16x16x4_f32(false, a, false, b3, (short)0, acc3, false, false);
    }

    float* crow = C + row0 * HID;
#pragma unroll
    for (int v = 0; v < 8; ++v) {
        const long long rr = (long long)(v + half * 8) * HID + l16;
        crow[rr +  0] = acc0[v];
        crow[rr + 16] = acc1[v];
        crow[rr + 32] = acc2[v];
        crow[rr + 48] = acc3[v];
    }
}

// ---------------- aggregation ----------------
// Self-loop term: out[i, :] = H[i, :] * dis[i]^2   (zero-initializes out; float4 wide)
__global__ void k_selfloop_init(const float4* __restrict__ H, const float* __restrict__ dis,
                                float4* __restrict__ out, long long total4) {
    long long t = (long long)blockIdx.x * blockDim.x + threadIdx.x;
    if (t >= total4) return;
    const int i = (int)(t >> 4);                     // 16 float4 per row of 64
    const float d2 = dis[i] * dis[i];
    float4 h = H[t];
    h.x *= d2; h.y *= d2; h.z *= d2; h.w *= d2;
    out[t] = h;
}

// Edge term: out[dst, f] += H[src, f] * norm[e]  — 2 features per thread, L2 atomics
__global__ void k_scatter(const float* __restrict__ H, const int* __restrict__ src,
                          const int* __restrict__ dst, const float* __restrict__ norm,
                          float* __restrict__ out, long long total32) {
    long long t = (long long)blockIdx.x * blockDim.x + threadIdx.x;
    if (t >= total32) return;
    const int e  = (int)(t >> 5);                    // 32 threads per edge
    const int f2 = (int)(t & 31) << 1;               // feature pair
    const int s = src[e];
    const int d = dst[e];
    const float w = norm[e];
    const v2f h = *(const v2f*)(H + (long long)s * HID + f2);
    float* o = out + (long long)d * HID + f2;
    atomicAdd(o + 0, h.x * w);
    atomicAdd(o + 1, h.y * w);
}

// a = relu(a + b), float4 wide
__global__ void k_bias_relu(float4* __restrict__ a, const float4* __restrict__ b, long long total4) {
    long long t = (long long)blockIdx.x * blockDim.x + threadIdx.x;
    if (t >= total4) return;
    const float4 bb = b[t & 15];                     // 16 float4 per 64-wide bias
    float4 v = a[t];
    v.x = fmaxf(v.x + bb.x, 0.0f);
    v.y = fmaxf(v.y + bb.y, 0.0f);
    v.z = fmaxf(v.z + bb.z, 0.0f);
    v.w = fmaxf(v.w + bb.w, 0.0f);
    a[t] = v;
}

// ---------------- head: logits = A @ W3 + b3; log_softmax over 2 classes ----------------
__global__ void k_head(const float* __restrict__ A, const float* __restrict__ W3,
                       const float* __restrict__ b3, float* __restrict__ out, int n) {
    const int i = blockIdx.x * blockDim.x + threadIdx.x;
    if (i >= n) return;
    const float* row = A + (long long)i * HID;
    float z0 = b3[0], z1 = b3[1];
#pragma unroll
    for (int k = 0; k < HID; ++k) {
        const float h = row[k];
        z0 += h * W3[2 * k];
        z1 += h * W3[2 * k + 1];
    }
    const float m   = fmaxf(z0, z1);
    const float lse = m + logf(expf(z0 - m) + expf(z1 - m));
    out[2 * i]     = z0 - lse;
    out[2 * i + 1] = z1 - lse;
}

extern "C" void kernel_launch(void* const* d_in, const int* in_sizes, int n_in,
                              void* d_out, int out_size, void* d_ws, size_t ws_size,
                              hipStream_t stream) {
    const float* x  = (const float*)d_in[0];
    const int*   ei = (const int*)d_in[1];   // [2, E] flat: src then dst
    const float* W1 = (const float*)d_in[2];
    const float* b1 = (const float*)d_in[3];
    const float* W2 = (const float*)d_in[4];
    const float* b2 = (const float*)d_in[5];
    const float* W3 = (const float*)d_in[6];
    const float* b3 = (const float*)d_in[7];
    float* out = (float*)d_out;

    const int N = N_NODES, E = E_EDGES;
    const int* src = ei;
    const int* dst = ei + E;

    // workspace: dis [N] | norm [E] | H [N*64] | A [N*64]  (~58 MB total)
    float* dis  = (float*)d_ws;
    float* norm = dis + N;
    float* H    = norm + E;
    float* A    = H + (size_t)N * HID;

    const long long ntot4  = (long long)N * HID / 4;   // 1.6M float4
    const long long etot32 = (long long)E * 32;        // 51.2M (2 feats/thread)
    const int nblk   = (N + 255) / 256;
    const int eblk   = (E + 255) / 256;
    const int nt4blk = (int)((ntot4 + 255) / 256);
    const int e32blk = (int)((etot32 + 255) / 256);
    const int gemm_blk = (int)(((long long)(N / 16) * 32 + 255) / 256);  // 16 rows/wave

    // degrees -> deg^{-1/2} -> per-edge norm
    k_deg_init    <<<nblk, 256, 0, stream>>>(dis, N);
    k_deg_accum   <<<eblk, 256, 0, stream>>>(dst, dis, E);
    k_deg_finalize<<<nblk, 256, 0, stream>>>(dis, N);
    k_edge_norm   <<<eblk, 256, 0, stream>>>(src, dst, dis, norm, E);

    // layer 1: H = x @ W1 ; A = aggregate(H) ; A = relu(A + b1)
    k_gemm_wmma    <<<gemm_blk, 256, 0, stream>>>(x, W1, H, N, IN_DIM);
    k_selfloop_init<<<nt4blk, 256, 0, stream>>>((const float4*)H, dis, (float4*)A, ntot4);
    k_scatter      <<<e32blk, 256, 0, stream>>>(H, src, dst, norm, A, etot32);
    k_bias_relu    <<<nt4blk, 256, 0, stream>>>((float4*)A, (const float4*)b1, ntot4);

    // layer 2: H = A @ W2 ; A = aggregate(H) ; A = relu(A + b2)
    k_gemm_wmma    <<<gemm_blk, 256, 0, stream>>>(A, W2, H, N, HID);
    k_selfloop_init<<<nt4blk, 256, 0, stream>>>((const float4*)H, dis, (float4*)A, ntot4);
    k_scatter      <<<e32blk, 256, 0, stream>>>(H, src, dst, norm, A, etot32);
    k_bias_relu    <<<nt4blk, 256, 0, stream>>>((float4*)A, (const float4*)b2, ntot4);

    // head
    k_head<<<nblk, 256, 0, stream>>>(A, W3, b3, out, N);
}